// SwitchLinear_34660386079265
// MI455X (gfx1250) — compile-verified
//
#include <hip/hip_runtime.h>

// SwitchLinear (per-sample channel-selected linear) for MI455X / gfx1250.
//
// Roofline: 1.07 GFLOP vs 24 MB minimum traffic -> ~1 us memory-bound at
// 23.3 TB/s. Use exact-f32 V_WMMA_F32_16X16X4_F32 (matches f32 reference,
// matrix pipe has huge headroom at this intensity).
//
// Pipeline (all on `stream`, all in d_ws):
//   1. zero channel counts
//   2. histogram channels + init perm to -1
//   3. 16-aligned exclusive scan of counts -> bucket offsets (+cursors)
//   4. scatter: perm[slot] = sample index (counting sort, rows only)
//   5. transpose weight [O,I,C] -> w_t[C][O][I] (LDS-tiled; makes B-fragment
//      loads contiguous b64 instead of one 128B line per 4B element)
//   6. grouped GEMM: per (channel, 32-row tile, 128-col tile) workgroup,
//      8 waves; X tile staged in LDS (stride 260 -> conflict-free ds_load_b64),
//      each wave: M=32 x N=16, K-loop step 4, 2 WMMAs/iter (dual acc chains).

typedef __attribute__((ext_vector_type(2))) float v2f;
typedef __attribute__((ext_vector_type(4))) float v4f;
typedef __attribute__((ext_vector_type(8))) float v8f;

#define B_N   8192
#define I_DIM 256
#define O_DIM 256
#define C_DIM 32
#define PAD_B (B_N + C_DIM * 16) // 8704 padded slots max

// ---------------- workspace layout (bytes) ----------------
// 0      : counts  int[32]
// 128    : offsets int[32]
// 256    : cursor  int[32]
// 512    : perm    int[8704]
// 65536  : w_t     float[32*256*256]  (8 MB)

__global__ void k_init_counts(int* __restrict__ counts) {
  if (threadIdx.x < C_DIM) counts[threadIdx.x] = 0;
}

__global__ void k_count(const int* __restrict__ idx, int* __restrict__ counts,
                        int* __restrict__ perm) {
  int t = blockIdx.x * 256 + threadIdx.x;
  if (t < PAD_B) perm[t] = -1;
  if (t < B_N) atomicAdd(&counts[idx[t]], 1);
}

__global__ void k_scan(const int* __restrict__ counts, int* __restrict__ offsets,
                       int* __restrict__ cursor) {
  if (threadIdx.x == 0) {
    int acc = 0;
    for (int c = 0; c < C_DIM; ++c) {
      offsets[c] = acc;
      cursor[c]  = acc;
      acc += (counts[c] + 15) & ~15; // pad each bucket to multiple of 16
    }
  }
}

__global__ void k_scatter(const int* __restrict__ idx, int* __restrict__ cursor,
                          int* __restrict__ perm) {
  int b = blockIdx.x * 256 + threadIdx.x;
  if (b < B_N) {
    int c   = idx[b];
    int pos = atomicAdd(&cursor[c], 1);
    perm[pos] = b; // slot order is nondeterministic; output values are not
  }
}

// weight is a [65536, 32] row-major matrix (rows = o*256+i, cols = c).
// Produce w_t = transpose: [32, 65536]. LDS tile 64 rows x 32 cols.
__global__ void k_transpose(const float* __restrict__ w, float* __restrict__ w_t) {
  __shared__ float tile[64][33]; // +1 pad: conflict-free write phase
  const int oiBase = blockIdx.x * 64;
  const int t = threadIdx.x;
#pragma unroll
  for (int j = 0; j < 8; ++j) { // coalesced read: consecutive t -> consecutive addr
    int lin = j * 256 + t;
    int cc  = lin & 31;
    int row = lin >> 5; // 0..63
    tile[row][cc] = w[(oiBase + row) * C_DIM + cc];
  }
  __syncthreads();
#pragma unroll
  for (int j = 0; j < 8; ++j) { // coalesced-ish write: 64-wide runs per channel
    int lin = j * 256 + t;
    int oi  = lin & 63;
    int cc  = lin >> 6; // 0..31
    w_t[cc * (O_DIM * I_DIM) + oiBase + oi] = tile[oi][cc];
  }
}

// Grouped GEMM. Block = 256 threads = 8 waves. Workgroup tile: M=32, N=128.
// grid = (O/128=2, ceil(B/32)=256, C=32); blocks beyond a bucket exit uniformly.
__launch_bounds__(256)
__global__ void k_gemm(const float* __restrict__ x, const float* __restrict__ w_t,
                       const float* __restrict__ bias,
                       const int* __restrict__ counts,
                       const int* __restrict__ offsets,
                       const int* __restrict__ perm, float* __restrict__ out) {
  const int c      = blockIdx.z;
  const int cnt    = counts[c];
  const int padded = (cnt + 15) & ~15;
  const int mBase  = blockIdx.y * 32;
  if (mBase >= padded) return;                   // uniform: EXEC stays all-ones
  const bool hasHi = (mBase + 16) < padded;      // upper 16-row subtile in-bucket?
  const int  base  = offsets[c];

  __shared__ float Xs[32][260]; // stride 260: (row*260+k)%64 = (4*row+k)%64 -> no conflicts
  __shared__ int   permS[32];

  const int t = threadIdx.x;
  if (t < 32) {
    int g = -1;
    if (t < 16 || hasHi) g = perm[base + mBase + t];
    permS[t] = g;
  }
  __syncthreads();

  { // stage X tile: 32 rows x 256 K, gathered by perm (rows are 1KB contiguous)
    const int row     = t >> 3;        // 0..31
    const int colBase = (t & 7) * 32;  // 0..224
    const int g       = permS[row];
    const float* src  = x + (long)g * I_DIM + colBase;
#pragma unroll
    for (int j = 0; j < 8; ++j) {
      v4f v = {};
      if (g >= 0) v = *(const v4f*)(src + j * 4);
      *(v4f*)(&Xs[row][colBase + j * 4]) = v; // 16B aligned (260%4==0)
    }
  }
  __syncthreads();

  const int wv    = t >> 5;                    // wave 0..7 -> N slice
  const int lane  = t & 31;
  const int nBase = blockIdx.x * 128 + wv * 16;
  const int o     = nBase + (lane & 15);       // this lane's N column
  const int koff  = (lane < 16) ? 0 : 2;       // ISA A/B fragment K split
  const int mRow  = lane & 15;                 // ISA A fragment M row

  const float* wp = w_t + (long)c * (O_DIM * I_DIM) + (long)o * I_DIM;

  v8f acc0 = {}; // rows mBase+0..15
  v8f acc1 = {}; // rows mBase+16..31
#pragma unroll 4
  for (int k0 = 0; k0 < I_DIM; k0 += 4) {
    const int kc = k0 + koff;
    v2f a0 = *(const v2f*)(&Xs[mRow][kc]);      // ds_load_b64
    v2f a1 = *(const v2f*)(&Xs[16 + mRow][kc]); // ds_load_b64
    v2f b  = *(const v2f*)(wp + kc);            // global_load_b64, contiguous K
    acc0 = __builtin_amdgcn_wmma_f32_16x16x4_f32(false, a0, false, b, (short)0,
                                                 acc0, false, false);
    acc1 = __builtin_amdgcn_wmma_f32_16x16x4_f32(false, a1, false, b, (short)0,
                                                 acc1, false, false);
  }

  // C/D layout: VGPR v, lanes 0-15 -> M=v, lanes 16-31 -> M=v+8; N = lane%16.
  const float bval  = bias[o * C_DIM + c];
  const int   mHalf = (lane < 16) ? 0 : 8;
#pragma unroll
  for (int v = 0; v < 8; ++v) {
    const int g0 = permS[mHalf + v];
    if (g0 >= 0) out[(long)g0 * O_DIM + o] = acc0[v] + bval;
    const int g1 = permS[16 + mHalf + v];
    if (g1 >= 0) out[(long)g1 * O_DIM + o] = acc1[v] + bval;
  }
}

extern "C" void kernel_launch(void* const* d_in, const int* in_sizes, int n_in,
                              void* d_out, int out_size, void* d_ws, size_t ws_size,
                              hipStream_t stream) {
  const float* x    = (const float*)d_in[0];
  const int*   idx  = (const int*)d_in[1]; // harness delivers integers as int32
  const float* w    = (const float*)d_in[2];
  const float* bias = (const float*)d_in[3];
  float*       out  = (float*)d_out;

  char* ws      = (char*)d_ws;
  int*  counts  = (int*)(ws + 0);
  int*  offsets = (int*)(ws + 128);
  int*  cursor  = (int*)(ws + 256);
  int*  perm    = (int*)(ws + 512);
  float* w_t    = (float*)(ws + 65536); // 8 MB

  k_init_counts<<<1, 64, 0, stream>>>(counts);
  k_count<<<(PAD_B + 255) / 256, 256, 0, stream>>>(idx, counts, perm);
  k_scan<<<1, 32, 0, stream>>>(counts, offsets, cursor);
  k_scatter<<<(B_N + 255) / 256, 256, 0, stream>>>(idx, cursor, perm);
  k_transpose<<<(O_DIM * I_DIM) / 64, 256, 0, stream>>>(w, w_t);

  dim3 grid(O_DIM / 128, (B_N + 31) / 32, C_DIM);
  k_gemm<<<grid, 256, 0, stream>>>(x, w_t, bias, counts, offsets, perm, out);
}